// DetectionLoss_71098888618645
// MI455X (gfx1250) — compile-verified
//
#include <hip/hip_runtime.h>
#include <hip/hip_bf16.h>
#include <stdint.h>

// ---------------- problem constants (match reference) ----------------
#define SGRID   52
#define NBOX    2
#define NCLS    8
#define CH      (NBOX * 5 + NCLS)   // 18 channels per cell
#define BATCH_N 512
#define NGT     32
#define LAMBDA_COORD 5.0f
#define LAMBDA_NOOBJ 0.5f
#define EPSF    1e-6f

// workspace accumulators (floats): 0=coord 1=obj 2=cls 3=softplus_total 4=marked_sub
#define ACC_N   8

__device__ __forceinline__ float softplusf(float x) {
    // numerically stable softplus: max(x,0) + log1p(exp(-|x|))
    return fmaxf(x, 0.f) + log1pf(__expf(-fabsf(x)));
}
__device__ __forceinline__ float sigmoidf_(float x) {
    return 1.f / (1.f + __expf(-x));
}
__device__ __forceinline__ long long llmin_(long long a, long long b) { return a < b ? a : b; }

// ---------------- kernel 0: zero accumulators ----------------
__global__ void zero_kernel(float* acc) {
    if (threadIdx.x < ACC_N) acc[threadIdx.x] = 0.f;
}

// ---------------- kernel 1: per-GT gather pass ----------------
// One wave32 per batch image; lane n handles GT n (NGT == 32 == warpSize).
__global__ __launch_bounds__(32) void yolo_gt_kernel(
    const float* __restrict__ pred, const float* __restrict__ boxes,
    const int* __restrict__ labels, const unsigned char* __restrict__ valid,
    float* __restrict__ acc)
{
    const int b = blockIdx.x;
    const int n = threadIdx.x;

    const float* gb = boxes + ((size_t)b * NGT + n) * 4;
    const float cx = gb[0], cy = gb[1], w = gb[2], h = gb[3];
    const int   vld = valid[(size_t)b * NGT + n] ? 1 : 0;
    int lab = labels[(size_t)b * NGT + n];
    lab = min(max(lab, 0), NCLS - 1);

    const int gi = min(max((int)floorf(cy * (float)SGRID), 0), SGRID - 1);
    const int gj = min(max((int)floorf(cx * (float)SGRID), 0), SGRID - 1);

    const float* cell = pred + (((size_t)b * SGRID + gi) * SGRID + gj) * CH;
    float c[CH];
    #pragma unroll
    for (int i = 0; i < CH; ++i) c[i] = cell[i];

    // decode both boxes and IoU vs GT
    float iou[NBOX], px[NBOX], py[NBOX], pwv[NBOX], phv[NBOX];
    const float gx1 = cx - w * 0.5f, gy1 = cy - h * 0.5f;
    const float gx2 = cx + w * 0.5f, gy2 = cy + h * 0.5f;
    #pragma unroll
    for (int k = 0; k < NBOX; ++k) {
        px[k]  = (sigmoidf_(c[5*k+1]) + (float)gj) / (float)SGRID;
        py[k]  = (sigmoidf_(c[5*k+2]) + (float)gi) / (float)SGRID;
        pwv[k] = fminf(__expf(c[5*k+3]), 1.f);
        phv[k] = fminf(__expf(c[5*k+4]), 1.f);
        const float x1 = px[k] - pwv[k]*0.5f, y1 = py[k] - phv[k]*0.5f;
        const float x2 = px[k] + pwv[k]*0.5f, y2 = py[k] + phv[k]*0.5f;
        const float iw = fmaxf(fminf(x2, gx2) - fmaxf(x1, gx1), 0.f);
        const float ih = fmaxf(fminf(y2, gy2) - fmaxf(y1, gy1), 0.f);
        const float inter = iw * ih;
        const float uni = (x2-x1)*(y2-y1) + (gx2-gx1)*(gy2-gy1) - inter + EPSF;
        iou[k] = inter / uni;
    }
    const int   best = (iou[1] > iou[0]) ? 1 : 0;   // argmax: first max wins ties
    const float bx = best ? px[1]  : px[0];
    const float by = best ? py[1]  : py[0];
    const float bw = best ? pwv[1] : pwv[0];
    const float bh = best ? phv[1] : phv[0];
    const float bobj = best ? c[5] : c[0];

    const float dsw = sqrtf(bw + EPSF) - sqrtf(w + EPSF);
    const float dsh = sqrtf(bh + EPSF) - sqrtf(h + EPSF);
    const float coord = (bx-cx)*(bx-cx) + (by-cy)*(by-cy) + dsw*dsw + dsh*dsh;
    const float objl  = softplusf(-bobj);

    // log-softmax NLL over 8 classes (branch-free label pick, avoids scratch)
    float m = c[10];
    #pragma unroll
    for (int i = 11; i < CH; ++i) m = fmaxf(m, c[i]);
    float se = 0.f, lv = 0.f;
    #pragma unroll
    for (int i = 0; i < NCLS; ++i) {
        se += __expf(c[10+i] - m);
        lv = (i == lab) ? c[10+i] : lv;
    }
    const float nll = (m + __logf(se)) - lv;

    const float fv = (float)vld;
    float coordS = coord * fv, objS = objl * fv, clsS = nll * fv;

    // Deduplicate marked cells within this image: only the FIRST valid GT that
    // maps to a cell subtracts that cell's objectness softplus from the
    // "all cells" total computed by the streaming kernel.
    const int key = gi * SGRID + gj;
    bool first = (vld != 0);
    for (int j = 0; j < NGT - 1; ++j) {          // uniform trip-count for shfl
        const int kj = __shfl(key, j);
        const int vj = __shfl(vld, j);
        if (j < n && vj && kj == key) first = false;
    }
    float subS = (vld && first) ? (softplusf(c[0]) + softplusf(c[5])) : 0.f;

    #pragma unroll
    for (int off = 16; off > 0; off >>= 1) {      // wave32 reduction
        coordS += __shfl_xor(coordS, off);
        objS   += __shfl_xor(objS,   off);
        clsS   += __shfl_xor(clsS,   off);
        subS   += __shfl_xor(subS,   off);
    }
    if (n == 0) {
        atomicAdd(&acc[0], coordS);
        atomicAdd(&acc[1], objS);
        atomicAdd(&acc[2], clsS);
        atomicAdd(&acc[4], subS);
    }
}

// ---------------- kernel 2: streaming noobj pass via TDM ----------------
#if __has_builtin(__builtin_amdgcn_tensor_load_to_lds)
#define HAVE_TDM 1
#else
#define HAVE_TDM 0
#endif

__device__ __forceinline__ void wait_tensorcnt0() {
#if __has_builtin(__builtin_amdgcn_s_wait_tensorcnt)
    __builtin_amdgcn_s_wait_tensorcnt(0);
#else
    asm volatile("s_wait_tensorcnt 0" ::: "memory");
#endif
}

#define ROWS 512   // cells per TDM tile
#define TB   256   // threads per block (8 waves)

#if HAVE_TDM
typedef unsigned int uint32x4_t __attribute__((ext_vector_type(4)));
typedef int          int32x8_t  __attribute__((ext_vector_type(8)));
typedef int          int32x4_t  __attribute__((ext_vector_type(4)));

// Build a 2-D D# (CDNA5 ISA) and issue TENSOR_LOAD_TO_LDS.
// Tensor: fp32 elements, dim0 = 18 channels (stride 18), dim1 = remaining cells.
// Tile: 6 x rows -> channels 0..5 of `rows` cells, packed contiguously in LDS.
// This toolchain (clang-23 / therock-10.0 headers) exposes the 6-arg builtin:
//   (uint32x4 g0, int32x8 g1, int32x4 g2, int32x4 g3, int32x8 extra, i32 cpol)
// count=1 in g0 -> only groups 0/1 are meaningful for our 2-D descriptor; the
// remaining groups are passed zero-filled.
__device__ __forceinline__ void tdm_issue(const float* gsrc, unsigned lds_byte_addr,
                                          unsigned long long rem_rows, int rows)
{
    const unsigned long long ga = (unsigned long long)(uintptr_t)gsrc;
    uint32x4_t g0;
    g0[0] = 1u;                                            // count=1, user descriptor
    g0[1] = lds_byte_addr;                                 // LDS dest (bytes)
    g0[2] = (unsigned)(ga & 0xFFFFFFFFu);                  // global_addr[31:0]
    g0[3] = (unsigned)((ga >> 32) & 0x1FFFFFFu)            // global_addr[56:32]
          | (2u << 30);                                    // type=2 ("image")
    const unsigned td1 = (rem_rows > 0xFFFFFFFFull) ? 0xFFFFFFFFu : (unsigned)rem_rows;
    int32x8_t g1;
    g1[0] = (int)(2u << 16);                               // data_size=2 (4 bytes)
    g1[1] = (int)(((unsigned)CH & 0xFFFFu) << 16);         // tensor_dim0 = 18 (bits 48..63)
    g1[2] = (int)((td1 & 0xFFFFu) << 16);                  // tensor_dim0 hi=0 | tensor_dim1 lo
    g1[3] = (int)((td1 >> 16) | (6u << 16));               // tensor_dim1 hi | tile_dim0=6
    g1[4] = (int)((unsigned)rows & 0xFFFFu);               // tile_dim1 = rows, tile_dim2=0
    g1[5] = (int)CH;                                       // tensor_dim0_stride = 18
    g1[6] = 0;                                             // stride hi / dim1_stride lo
    g1[7] = 0;
    const int32x4_t z4 = {0, 0, 0, 0};                     // groups 2/3 unused (2-D)
    const int32x8_t z8 = {0, 0, 0, 0, 0, 0, 0, 0};         // trailing group (unused)
    __builtin_amdgcn_tensor_load_to_lds(g0, g1, z4, z4, z8, 0);
}
#endif

__global__ __launch_bounds__(TB) void noobj_kernel(const float* __restrict__ pred,
                                                   float* __restrict__ acc)
{
    const long long total = (long long)BATCH_N * SGRID * SGRID;
    float sum = 0.f;
#if HAVE_TDM
    __shared__ float buf[2][ROWS * 6];                     // 24 KB double buffer
    const long long base0  = (long long)blockIdx.x * ROWS;
    const long long stride = (long long)gridDim.x * ROWS;
    if (base0 < total) {
        if (threadIdx.x == 0) {                            // TDM ignores EXEC: 1 issue/block
            const int r0 = (int)llmin_((long long)ROWS, total - base0);
            tdm_issue(pred + base0 * CH,
                      (unsigned)(uintptr_t)&buf[0][0],     // LDS aperture keeps offset in addr[31:0]
                      (unsigned long long)(total - base0), r0);
        }
        int par = 0;
        for (long long base = base0; base < total; base += stride, par ^= 1) {
            if (threadIdx.x == 0) wait_tensorcnt0();       // tile `base` landed in LDS
            __syncthreads();
            const long long nbase = base + stride;         // prefetch next tile (overlaps compute)
            if (nbase < total && threadIdx.x == 0) {
                const int rn = (int)llmin_((long long)ROWS, total - nbase);
                tdm_issue(pred + nbase * CH,
                          (unsigned)(uintptr_t)&buf[par ^ 1][0],
                          (unsigned long long)(total - nbase), rn);
            }
            const int rows = (int)llmin_((long long)ROWS, total - base);
            for (int r = threadIdx.x; r < rows; r += TB) { // conflict-free: 6r mod 64 distinct
                sum += softplusf(buf[par][r * 6 + 0]) + softplusf(buf[par][r * 6 + 5]);
            }
            __syncthreads();                               // buffer reuse fence
        }
    }
#else
    for (long long cidx = (long long)blockIdx.x * TB + threadIdx.x; cidx < total;
         cidx += (long long)gridDim.x * TB) {
        const float* p = pred + cidx * CH;
        sum += softplusf(p[0]) + softplusf(p[5]);
    }
#endif
    #pragma unroll
    for (int off = 16; off > 0; off >>= 1) sum += __shfl_xor(sum, off);
    if ((threadIdx.x & 31) == 0) atomicAdd(&acc[3], sum);
}

// ---------------- kernel 3: finalize 5 outputs ----------------
__global__ void finalize_kernel(const float* __restrict__ acc, float* __restrict__ out) {
    if (threadIdx.x == 0) {
        const float coord = acc[0], obj = acc[1], cls = acc[2];
        const float noobj = acc[3] - acc[4];   // total softplus minus marked cells
        const float inv = 1.0f / (float)BATCH_N;
        out[0] = (LAMBDA_COORD * coord + obj + LAMBDA_NOOBJ * noobj + cls) * inv;
        out[1] = coord * inv;
        out[2] = obj * inv;
        out[3] = noobj * inv;
        out[4] = cls * inv;
    }
}

// ---------------- launch ----------------
extern "C" void kernel_launch(void* const* d_in, const int* in_sizes, int n_in,
                              void* d_out, int out_size, void* d_ws, size_t ws_size,
                              hipStream_t stream) {
    const float*         pred   = (const float*)d_in[0];
    const float*         boxes  = (const float*)d_in[1];
    const int*           labels = (const int*)d_in[2];
    const unsigned char* valid  = (const unsigned char*)d_in[3];
    float* acc = (float*)d_ws;
    float* out = (float*)d_out;

    zero_kernel<<<dim3(1), dim3(32), 0, stream>>>(acc);
    yolo_gt_kernel<<<dim3(BATCH_N), dim3(32), 0, stream>>>(pred, boxes, labels, valid, acc);
    noobj_kernel<<<dim3(512), dim3(TB), 0, stream>>>(pred, acc);
    finalize_kernel<<<dim3(1), dim3(32), 0, stream>>>(acc, out);
}